// TemporalSelfAttention_PyTorch_41747082117463
// MI455X (gfx1250) — compile-verified
//
#include <hip/hip_runtime.h>
#include <hip/hip_bf16.h>

// ---------------------------------------------------------------------------
// TemporalSelfAttention (deformable attention) for MI455X / gfx1250
//  - GEMMs on V_WMMA_F32_16X16X4_F32 (fp32 matrix pipe, wave32)
//  - B panel staged into LDS via the Tensor Data Mover (tensor_load_to_lds,
//    TENSORcnt) when available; cooperative loads otherwise
// ---------------------------------------------------------------------------

typedef float v2f __attribute__((ext_vector_type(2)));
typedef float v8f __attribute__((ext_vector_type(8)));
typedef unsigned int v4u __attribute__((ext_vector_type(4)));
typedef int v4i __attribute__((ext_vector_type(4)));
typedef int v8i __attribute__((ext_vector_type(8)));

#if defined(__AMDGCN__) && __has_builtin(__builtin_amdgcn_tensor_load_to_lds)
#define USE_TDM 1
#else
#define USE_TDM 0
#endif

#define MAX_K 512  // largest GEMM K in this model (Din)

// ---------------------------------------------------------------------------
// 1) query_concat[b,q,0:256]=value_flat[b,q,:], [256:512]=query[b,q,:]
// ---------------------------------------------------------------------------
__global__ void concat_qv_kernel(const float* __restrict__ value,
                                 const float* __restrict__ query,
                                 float* __restrict__ qcat, int nq) {
  const int n = 2 * nq * 512;
  for (int i = blockIdx.x * blockDim.x + threadIdx.x; i < n;
       i += gridDim.x * blockDim.x) {
    const int k   = i & 511;
    const int row = i >> 9;  // b*nq + q
    qcat[i] = (k < 256) ? value[(size_t)row * 256 + k]
                        : query[(size_t)row * 256 + (k - 256)];
  }
}

// ---------------------------------------------------------------------------
// 2) C[M,N] = A[M,K] @ B[K,N] + bias[N] (+ resid[M,N])
//    Block: 8 waves sharing one 16-column panel (n0); B[K x 16] staged in LDS
//    (TDM descriptor per CDNA5 ISA). Each wave owns one
//    16x16 output tile and runs the K-loop on V_WMMA_F32_16X16X4_F32.
//    A fragment: lane L holds A[m0 + L%16][k0 + 2*(L/16) + {0,1}]
//    B fragment: lane L holds B[k0 + 2*(L/16) + {0,1}][n0 + L%16]
//    C/D: VGPR j on lane L = C[m0 + j + 8*(L/16)][n0 + L%16]
// ---------------------------------------------------------------------------
__global__ void wmma_gemm_f32_kernel(const float* __restrict__ A,
                                     const float* __restrict__ B,
                                     const float* __restrict__ bias,
                                     const float* __restrict__ resid,
                                     float* __restrict__ C,
                                     int M, int K, int N) {
  __shared__ float sB[MAX_K * 16];  // 32 KB of the 320 KB WGP LDS

  const int lane = threadIdx.x & 31;
  const int wave = threadIdx.x >> 5;
  const int n0   = blockIdx.x * 16;

#if USE_TDM
  if (wave == 0) {
    // ---- Tensor DMA descriptor (D#): 2D tile, B[0:K, n0:n0+16] -> sB ----
    const unsigned long long gaddr =
        (unsigned long long)(uintptr_t)(B + n0);          // tile start (bytes)
    const unsigned int ldsAddr =
        (unsigned int)(uintptr_t)&sB[0];                   // LDS byte offset
    v4u g0;
    g0.x = 1u;                                             // count=1, user D#
    g0.y = ldsAddr;                                        // lds_addr
    g0.z = (unsigned int)(gaddr & 0xFFFFFFFFu);            // global_addr[31:0]
    g0.w = (unsigned int)((gaddr >> 32) & 0x01FFFFFFu)     // global_addr[56:32]
           | (2u << 30);                                   // type = 2 ("image")
    v8i g1;
    g1[0] = (int)(2u << 16);           // wg_mask=0, data_size=4B, no flags
    g1[1] = (int)(16u << 16);          // tensor_dim0 = 16 (tile-sized, in-bounds)
    g1[2] = (int)(((unsigned)K & 0xFFFFu) << 16);  // tensor_dim1 = K
    g1[3] = (int)(16u << 16);          // tile_dim0 = 16
    g1[4] = (int)((unsigned)K);        // tile_dim1 = K, tile_dim2 = 0
    g1[5] = (int)((unsigned)N);        // tensor_dim0_stride[31:0] = N
    g1[6] = 0;                         // stride0[47:32]=0, stride1 lo=0
    g1[7] = 0;
    v4i gz = {0, 0, 0, 0};
#if __clang_major__ >= 23
    v8i gz8 = {0, 0, 0, 0, 0, 0, 0, 0};
    __builtin_amdgcn_tensor_load_to_lds(g0, g1, gz, gz, gz8, 0);
#else
    __builtin_amdgcn_tensor_load_to_lds(g0, g1, gz, gz, 0);
#endif
    __builtin_amdgcn_s_wait_tensorcnt(0);
  }
#else
  // Cooperative fallback: 256 threads copy the K x 16 panel.
  for (int i = threadIdx.x; i < K * 16; i += blockDim.x)
    sB[i] = B[(size_t)(i >> 4) * N + n0 + (i & 15)];
#endif
  __syncthreads();

  const int tileM = blockIdx.y * (blockDim.x >> 5) + wave;
  const int m0    = tileM * 16;
  if (m0 >= M) return;  // wave-uniform; EXEC stays all-ones for live waves

  const int col16 = lane & 15;
  const int kSel  = (lane >> 4) << 1;  // 0 or 2

  // Clamp A row so OOB tile rows read valid memory; store predicated below.
  const int arow = (m0 + col16 < M) ? (m0 + col16) : (M - 1);
  const float* __restrict__ aptr = A + (size_t)arow * K + kSel;
  const float* __restrict__ bl   = &sB[col16];

  v8f acc = {};
#pragma unroll 4
  for (int k0 = 0; k0 < K; k0 += 4) {
    __builtin_prefetch(aptr + k0 + 64, 0, 1);  // global_prefetch_b8
    const v2f a = *(const v2f*)(aptr + k0);    // A[arow][k0+kSel, +1]
    v2f b;
    b.x = bl[(k0 + kSel) * 16];                // ds_load from staged panel
    b.y = bl[(k0 + kSel + 1) * 16];
    acc = __builtin_amdgcn_wmma_f32_16x16x4_f32(
        /*neg_a=*/false, a, /*neg_b=*/false, b,
        /*c_mod=*/(short)0, acc, /*reuse_a=*/false, /*reuse_b=*/false);
  }

  const int   rbase = m0 + ((lane >> 4) << 3);
  const float bv    = bias[n0 + col16];
#pragma unroll
  for (int j = 0; j < 8; ++j) {
    const int row = rbase + j;
    if (row < M) {
      float o = acc[j] + bv;
      if (resid) o += resid[(size_t)row * N + n0 + col16];
      C[(size_t)row * N + n0 + col16] = o;
    }
  }
}

// ---------------------------------------------------------------------------
// 3) softmax over contiguous groups of 16 (the L*P axis)
// ---------------------------------------------------------------------------
__global__ void softmax16_kernel(float* __restrict__ aw, int ngroups) {
  const int g = blockIdx.x * blockDim.x + threadIdx.x;
  if (g >= ngroups) return;
  float* p = aw + (size_t)g * 16;
  float v[16];
  float m = -3.0e38f;
#pragma unroll
  for (int i = 0; i < 16; ++i) { v[i] = p[i]; m = fmaxf(m, v[i]); }
  float s = 0.f;
#pragma unroll
  for (int i = 0; i < 16; ++i) { v[i] = __expf(v[i] - m); s += v[i]; }
  const float inv = 1.0f / s;
#pragma unroll
  for (int i = 0; i < 16; ++i) p[i] = v[i] * inv;
}

// ---------------------------------------------------------------------------
// 4) bilinear sample + attention-weighted sum + mean over nbq.
//    One wave per (b, q, h); lane = channel c (hd = 32 = wave32, so every
//    corner gather is a fully coalesced 128B line). 4 heads per block.
// ---------------------------------------------------------------------------
__global__ void deform_sample_kernel(const float* __restrict__ vproj,  // [BQ,nq,H,hd]
                                     const float* __restrict__ off,    // [bs,nq,512]
                                     const float* __restrict__ aw,     // [bs,nq,256]
                                     const float* __restrict__ refpts, // [bs,nq,L,2]
                                     const int*   __restrict__ shapes, // [L,2] (h,w)
                                     float* __restrict__ pre,          // [bs,nq,256]
                                     int nq) {
  const int c = threadIdx.x;                       // 0..31 channel
  const int h = blockIdx.y * blockDim.y + threadIdx.y;
  const int q = blockIdx.x;
  const int b = blockIdx.z;

  int hl[4], wl[4], start[4];
  int s = 0;
#pragma unroll
  for (int l = 0; l < 4; ++l) {
    hl[l] = shapes[2 * l];
    wl[l] = shapes[2 * l + 1];
    start[l] = s;
    s += hl[l] * wl[l];
  }

  const size_t rowq = (size_t)b * nq + q;
  const float* __restrict__ offr = off + rowq * 512;
  const float* __restrict__ awr  = aw + rowq * 256;
  const float* __restrict__ refr = refpts + rowq * 8;  // L*2

  float accsum = 0.f;
  for (int nb = 0; nb < 2; ++nb) {
    const int bq = b * 2 + nb;
    const float* __restrict__ vb = vproj + (size_t)bq * nq * 256 + h * 32 + c;
    for (int l = 0; l < 4; ++l) {
      const float Wf = (float)wl[l], Hf = (float)hl[l];
      const float rx = refr[2 * l], ry = refr[2 * l + 1];
#pragma unroll
      for (int p = 0; p < 4; ++p) {
        const int base = (((h * 2 + nb) * 4 + l) * 4 + p);
        const float ox = offr[base * 2];
        const float oy = offr[base * 2 + 1];
        const float w  = awr[base];
        // grid = 2*(ref + off/norm) - 1; unnormalize (align_corners=False)
        const float gx = 2.f * (rx + ox / Wf) - 1.f;
        const float gy = 2.f * (ry + oy / Hf) - 1.f;
        const float x = (gx + 1.f) * (Wf * 0.5f) - 0.5f;
        const float y = (gy + 1.f) * (Hf * 0.5f) - 0.5f;
        const float x0f = floorf(x), y0f = floorf(y);
        const int   x0 = (int)x0f, y0 = (int)y0f;
        const float wx = x - x0f, wy = y - y0f;
        float sval = 0.f;
#pragma unroll
        for (int dy = 0; dy < 2; ++dy) {
          const int iy = y0 + dy;
          if (iy < 0 || iy >= hl[l]) continue;
          const float fy = dy ? wy : (1.f - wy);
#pragma unroll
          for (int dx = 0; dx < 2; ++dx) {
            const int ix = x0 + dx;
            if (ix < 0 || ix >= wl[l]) continue;
            const float fx = dx ? wx : (1.f - wx);
            const size_t idx = (size_t)(start[l] + iy * wl[l] + ix) * 256;
            sval += fx * fy * vb[idx];  // zero-pad OOB via the guards above
          }
        }
        accsum += w * sval;
      }
    }
  }
  pre[rowq * 256 + h * 32 + c] = accsum * 0.5f;  // mean over nbq = 2
}

// ---------------------------------------------------------------------------
// launcher
// ---------------------------------------------------------------------------
static inline void launch_gemm(const float* A, const float* B, const float* bias,
                               const float* resid, float* C, int M, int K, int N,
                               hipStream_t stream) {
  const int wavesPerBlock = 8;
  const int mtiles = (M + 15) / 16;
  dim3 grid(N / 16, (mtiles + wavesPerBlock - 1) / wavesPerBlock);
  wmma_gemm_f32_kernel<<<grid, wavesPerBlock * 32, 0, stream>>>(
      A, B, bias, resid, C, M, K, N);
}

extern "C" void kernel_launch(void* const* d_in, const int* in_sizes, int n_in,
                              void* d_out, int out_size, void* d_ws, size_t ws_size,
                              hipStream_t stream) {
  const float* query  = (const float*)d_in[0];   // [bs, nq, 256]
  const float* value  = (const float*)d_in[1];   // [BQ, nq, 8, 32] == [BQ,nq,256]
  const float* refpts = (const float*)d_in[2];   // [bs, nq, 4, 2]
  const int*   shapes = (const int*)d_in[3];     // [4, 2]
  const float* W_samp = (const float*)d_in[4];   // [512, 512]
  const float* b_samp = (const float*)d_in[5];   // [512]
  const float* W_attn = (const float*)d_in[6];   // [512, 256]
  const float* b_attn = (const float*)d_in[7];   // [256]
  const float* W_val  = (const float*)d_in[8];   // [256, 256]
  const float* b_val  = (const float*)d_in[9];   // [256]
  const float* W_out  = (const float*)d_in[10];  // [256, 256]
  const float* b_out  = (const float*)d_in[11];  // [256]
  float* out = (float*)d_out;                    // [bs, nq, 256]

  const int bs = 2, nbq = 2, H = 8, D = 256;
  const int nq = in_sizes[0] / (bs * D);  // 3343
  const int BQ = bs * nbq;
  const int Din = 2 * D;  // 512

  float* ws = (float*)d_ws;
  size_t o = 0;
  float* qcat = ws + o; o += (size_t)bs * nq * Din;  // [2*nq, 512]
  float* vprj = ws + o; o += (size_t)BQ * nq * D;    // [4*nq, 256]
  float* offb = ws + o; o += (size_t)bs * nq * Din;  // [2*nq, 512]
  float* awb  = ws + o; o += (size_t)bs * nq * D;    // [2*nq, 256]
  float* pre  = ws + o; o += (size_t)bs * nq * D;    // [2*nq, 256]
  (void)ws_size; (void)n_in; (void)out_size;

  // 1) query_concat
  {
    const int n = bs * nq * Din;
    int blocks = (n + 255) / 256;
    if (blocks > 1024) blocks = 1024;
    concat_qv_kernel<<<blocks, 256, 0, stream>>>(value, query, qcat, nq);
  }

  // 2) value projection: v = value_flat @ W_val + b_val      [13372 x 256]
  launch_gemm(value, W_val, b_val, nullptr, vprj, BQ * nq, D, D, stream);

  // 3) sampling offsets: off = qcat @ W_samp + b_samp        [6686 x 512]
  launch_gemm(qcat, W_samp, b_samp, nullptr, offb, bs * nq, Din, Din, stream);

  // 4) attention logits: aw = qcat @ W_attn + b_attn         [6686 x 256]
  launch_gemm(qcat, W_attn, b_attn, nullptr, awb, bs * nq, Din, D, stream);

  // 5) softmax over L*P = 16 for each (b, q, h, nb)
  {
    const int ngroups = bs * nq * H * nbq;
    softmax16_kernel<<<(ngroups + 255) / 256, 256, 0, stream>>>(awb, ngroups);
  }

  // 6) deformable bilinear sampling + weighted sum + mean over nbq
  deform_sample_kernel<<<dim3(nq, H / 4, bs), dim3(32, 4), 0, stream>>>(
      vprj, offb, awb, refpts, shapes, pre, nq);

  // 7) output projection + residual: out = pre @ W_out + b_out + query
  launch_gemm(pre, W_out, b_out, query, out, bs * nq, D, D, stream);
}